// CAM_Module_72430328481281
// MI455X (gfx1250) — compile-verified
//
#include <hip/hip_runtime.h>

typedef __attribute__((ext_vector_type(2))) float v2f;
typedef __attribute__((ext_vector_type(8))) float v8f;

#define B_   8
#define C_   512
#define N_   4096
#define TILE 64
#define KC   32
#define LDA  36   // KC + 4 pad: col reads hit 16 distinct banks (36m mod 64)
#define LDB3 72   // 64 + 8 pad: 2-row offset = 144 ≡ 16 mod 64 -> disjoint half-wave banks

// Low 32 bits of a generic pointer into LDS == LDS byte offset (aperture maps addr[31:0]).
__device__ __forceinline__ unsigned lds_off(const void* p) {
  return (unsigned)(reinterpret_cast<size_t>(p));
}

// Async DMA 32 bytes/thread straight into LDS (2 x B128). INST_OFFSET is added to
// both the LDS and global addresses, so one base register covers both halves.
// Tracked on ASYNCcnt (in-order completion).
__device__ __forceinline__ void async_tile32B(unsigned lds, const float* g) {
  asm volatile("global_load_async_to_lds_b128 %0, %1, off"
               :: "v"(lds), "v"(g) : "memory");
  asm volatile("global_load_async_to_lds_b128 %0, %1, off offset:16"
               :: "v"(lds), "v"(g) : "memory");
}

__device__ __forceinline__ void wait_async4() {
  asm volatile("s_wait_asynccnt 0x4" ::: "memory");
}
__device__ __forceinline__ void wait_async0() {
  asm volatile("s_wait_asynccnt 0x0" ::: "memory");
}

// ---------------------------------------------------------------------------
// Kernel 1: energy[b] = Q * Q^T   (Q = x[b] viewed as [C_, N_], fp32)
// Grid: (64, 8) = (8x8 C-tiles, B). Block: 256 threads = 8 waves.
// Double-buffered LDS fed by global_load_async_to_lds_b128; each wave computes
// a 16x32 sub-tile = two v8f accumulators via V_WMMA_F32_16X16X4_F32.
// Final K-chunk peeled so the steady-state loop is branch-free.
// ---------------------------------------------------------------------------
__global__ __launch_bounds__(256) void cam_energy(const float* __restrict__ x,
                                                  float* __restrict__ energy) {
  __shared__ float As[2][TILE][LDA];
  __shared__ float Bs[2][TILE][LDA];

  const int b  = blockIdx.y;
  const int ti = blockIdx.x >> 3;   // row tile of 64
  const int tj = blockIdx.x & 7;    // col tile of 64
  const float* Q = x + (size_t)b * C_ * N_;

  const int tid   = threadIdx.x;
  const int lane  = tid & 31;
  const int wave  = tid >> 5;        // 0..7
  const int strip = wave >> 1;       // 16-row strip (0..3)
  const int cblk  = (wave & 1) * 2;  // first of two 16-col blocks

  const int m     = lane & 15;          // M (for A) / N (for B) index
  const int khalf = (lane >> 4) << 1;   // lanes 0-15 -> K=0,1 ; 16-31 -> K=2,3

  v8f acc0 = {};
  v8f acc1 = {};

  const int lr = tid >> 2;         // load row 0..63
  const int lc = (tid & 3) << 3;   // load col 0,8,16,24

  const float* gA = Q + (size_t)(ti * TILE + lr) * N_ + lc;
  const float* gB = Q + (size_t)(tj * TILE + lr) * N_ + lc;
  const unsigned laA[2] = {lds_off(&As[0][lr][lc]), lds_off(&As[1][lr][lc])};
  const unsigned laB[2] = {lds_off(&Bs[0][lr][lc]), lds_off(&Bs[1][lr][lc])};

  auto mma_tile = [&](int s) {
#pragma unroll
    for (int kk = 0; kk < KC; kk += 4) {
      v2f a, f0, f1;
      a.x  = As[s][strip * 16 + m][kk + khalf];
      a.y  = As[s][strip * 16 + m][kk + khalf + 1];
      f0.x = Bs[s][cblk * 16 + m][kk + khalf];
      f0.y = Bs[s][cblk * 16 + m][kk + khalf + 1];
      f1.x = Bs[s][(cblk + 1) * 16 + m][kk + khalf];
      f1.y = Bs[s][(cblk + 1) * 16 + m][kk + khalf + 1];
      acc0 = __builtin_amdgcn_wmma_f32_16x16x4_f32(false, a, false, f0, (short)0, acc0, false, false);
      acc1 = __builtin_amdgcn_wmma_f32_16x16x4_f32(false, a, false, f1, (short)0, acc1, false, false);
    }
  };

  // prologue: stage 0 in flight (4 async B128 per thread)
  async_tile32B(laA[0], gA);
  async_tile32B(laB[0], gB);

  int cur = 0;
  for (int k0 = 0; k0 < N_ - KC; k0 += KC) {
    // buf cur^1 was last read in the previous iteration (trailing barrier protects it)
    async_tile32B(laA[cur ^ 1], gA + k0 + KC);
    async_tile32B(laB[cur ^ 1], gB + k0 + KC);
    wait_async4();            // cur's 4 done (in-order), next 4 still in flight
    __syncthreads();
    mma_tile(cur);
    __syncthreads();
    cur ^= 1;
  }
  // peeled final chunk: nothing left to issue
  wait_async0();
  __syncthreads();
  mma_tile(cur);

  // C/D layout: VGPR v -> row v (lanes 0-15) or v+8 (lanes 16-31), col = lane&15
  float* E = energy + (size_t)b * C_ * C_;
  const int irow0 = ti * TILE + strip * 16 + ((lane >> 4) << 3);
  const int jc0   = tj * TILE + cblk * 16 + m;
#pragma unroll
  for (int v = 0; v < 8; ++v) {
    E[(size_t)(irow0 + v) * C_ + jc0]      = acc0[v];
    E[(size_t)(irow0 + v) * C_ + jc0 + 16] = acc1[v];
  }
}

// ---------------------------------------------------------------------------
// Kernel 2: in-place reversed-max softmax per row.
// softmax(max_i - e_ij) stabilized == exp(min_i - e_ij) / sum_j exp(min_i - e_ij)
// One wave32 per 512-wide row. Grid 512, block 256 (8 rows/block).
// ---------------------------------------------------------------------------
__global__ __launch_bounds__(256) void cam_softmax(float* __restrict__ e) {
  const int row  = blockIdx.x * 8 + (threadIdx.x >> 5);
  const int lane = threadIdx.x & 31;
  float* r = e + (size_t)row * C_;

  float vals[16];
  float mn = 3.402823466e38f;
#pragma unroll
  for (int i = 0; i < 16; ++i) {
    vals[i] = r[lane + 32 * i];
    mn = fminf(mn, vals[i]);
  }
#pragma unroll
  for (int off = 16; off > 0; off >>= 1)
    mn = fminf(mn, __shfl_xor(mn, off, 32));

  float s = 0.f;
#pragma unroll
  for (int i = 0; i < 16; ++i) {
    vals[i] = __expf(mn - vals[i]);
    s += vals[i];
  }
#pragma unroll
  for (int off = 16; off > 0; off >>= 1)
    s += __shfl_xor(s, off, 32);

  const float inv = 1.0f / s;
#pragma unroll
  for (int i = 0; i < 16; ++i)
    r[lane + 32 * i] = vals[i] * inv;
}

// ---------------------------------------------------------------------------
// Kernel 3: out = gamma * (attn @ Q) + x.  M=512, N=4096, K=512 per batch.
// Grid: (64 n-tiles, 8 i-tiles, 8 batches). Block 256 = 8 waves, same tiling.
// Q re-read hits the 192MB L2 (x is only 64MB). Async-LDS double buffer,
// final K-chunk peeled.
// ---------------------------------------------------------------------------
__global__ __launch_bounds__(256) void cam_out(const float* __restrict__ attn,
                                               const float* __restrict__ x,
                                               const float* __restrict__ gamma,
                                               float* __restrict__ out) {
  __shared__ float As[2][TILE][LDA];   // attn tile: 64 (i) x 32 (k)
  __shared__ float Bs[2][KC][LDB3];    // Q tile:    32 (k) x 64 (n)

  const int b  = blockIdx.z;
  const int ti = blockIdx.y;
  const int tn = blockIdx.x;
  const float* A = attn + (size_t)b * C_ * C_;
  const float* Q = x + (size_t)b * C_ * N_;

  const int tid   = threadIdx.x;
  const int lane  = tid & 31;
  const int wave  = tid >> 5;
  const int strip = wave >> 1;
  const int cblk  = (wave & 1) * 2;

  const int m     = lane & 15;
  const int khalf = (lane >> 4) << 1;

  v8f acc0 = {};
  v8f acc1 = {};

  const int lrA = tid >> 2;        // 0..63
  const int lcA = (tid & 3) << 3;  // 0..24
  const int lrB = tid >> 3;        // 0..31
  const int lcB = (tid & 7) << 3;  // 0..56

  const float* gA = A + (size_t)(ti * TILE + lrA) * C_ + lcA;            // + k0
  const float* gB = Q + (size_t)lrB * N_ + tn * TILE + lcB;              // + k0*N_
  const unsigned laA[2] = {lds_off(&As[0][lrA][lcA]), lds_off(&As[1][lrA][lcA])};
  const unsigned laB[2] = {lds_off(&Bs[0][lrB][lcB]), lds_off(&Bs[1][lrB][lcB])};

  auto mma_tile = [&](int s) {
#pragma unroll
    for (int kk = 0; kk < KC; kk += 4) {
      v2f a, f0, f1;
      a.x  = As[s][strip * 16 + m][kk + khalf];
      a.y  = As[s][strip * 16 + m][kk + khalf + 1];
      f0.x = Bs[s][kk + khalf][cblk * 16 + m];
      f0.y = Bs[s][kk + khalf + 1][cblk * 16 + m];
      f1.x = Bs[s][kk + khalf][(cblk + 1) * 16 + m];
      f1.y = Bs[s][kk + khalf + 1][(cblk + 1) * 16 + m];
      acc0 = __builtin_amdgcn_wmma_f32_16x16x4_f32(false, a, false, f0, (short)0, acc0, false, false);
      acc1 = __builtin_amdgcn_wmma_f32_16x16x4_f32(false, a, false, f1, (short)0, acc1, false, false);
    }
  };

  async_tile32B(laA[0], gA);
  async_tile32B(laB[0], gB);

  int cur = 0;
  for (int k0 = 0; k0 < C_ - KC; k0 += KC) {
    async_tile32B(laA[cur ^ 1], gA + k0 + KC);
    async_tile32B(laB[cur ^ 1], gB + (size_t)(k0 + KC) * N_);
    wait_async4();
    __syncthreads();
    mma_tile(cur);
    __syncthreads();
    cur ^= 1;
  }
  wait_async0();
  __syncthreads();
  mma_tile(cur);

  const float g = gamma[0];
  const int irow0 = ti * TILE + strip * 16 + ((lane >> 4) << 3);
  const int nc0   = tn * TILE + cblk * 16 + m;
#pragma unroll
  for (int v = 0; v < 8; ++v) {
    const size_t i0 = ((size_t)b * C_ + irow0 + v) * N_ + nc0;
    out[i0]      = g * acc0[v] + x[i0];
    out[i0 + 16] = g * acc1[v] + x[i0 + 16];
  }
}

// ---------------------------------------------------------------------------
extern "C" void kernel_launch(void* const* d_in, const int* in_sizes, int n_in,
                              void* d_out, int out_size, void* d_ws, size_t ws_size,
                              hipStream_t stream) {
  const float* x     = (const float*)d_in[0];
  const float* gamma = (const float*)d_in[1];
  float* out         = (float*)d_out;
  float* energy      = (float*)d_ws;   // B_*C_*C_ floats = 8 MiB scratch (in-place softmax)

  (void)in_sizes; (void)n_in; (void)out_size; (void)ws_size;

  cam_energy<<<dim3(64, B_), 256, 0, stream>>>(x, energy);
  cam_softmax<<<dim3((B_ * C_) / 8), 256, 0, stream>>>(energy);
  cam_out<<<dim3(N_ / TILE, C_ / TILE, B_), 256, 0, stream>>>(energy, x, gamma, out);
}